// MACCell_9371618640099
// MI455X (gfx1250) — compile-verified
//
#include <hip/hip_runtime.h>
#include <hip/hip_bf16.h>
#include <stdint.h>

typedef __attribute__((ext_vector_type(2))) float v2f;
typedef __attribute__((ext_vector_type(8))) float v8f;
typedef __attribute__((ext_vector_type(4))) unsigned int v4u;
typedef __attribute__((ext_vector_type(4))) int v4i;
typedef __attribute__((ext_vector_type(8))) int v8i;

#define D 512
#define NHW 196

#if defined(__has_builtin)
#if __has_builtin(__builtin_amdgcn_tensor_load_to_lds) && \
    __has_builtin(__builtin_amdgcn_s_wait_tensorcnt)
#define USE_TDM 1
#endif
#endif

// ---------------------------------------------------------------------------
// Generic GEMM: C[M,N] = A[M,K] @ W[N,K]^T (+ bias) (+= if accumulate)
// grid = (N/16, M/16), block = 32 (one wave per 16x16 tile), fp32 WMMA.
// ---------------------------------------------------------------------------
__global__ __launch_bounds__(32) void gemm_xwT(
    const float* __restrict__ A, const float* __restrict__ W,
    const float* __restrict__ bias, float* __restrict__ C,
    int N, int K, int lda, int ldw, int accumulate)
{
    const int nBase = blockIdx.x * 16;
    const int mBase = blockIdx.y * 16;
    const int lane  = threadIdx.x;
    const int r16   = lane & 15;
    const int kOff  = (lane >> 4) * 2;
    const int rowHi = (lane >> 4) * 8;

    const float* Ap = A + (size_t)(mBase + r16) * lda + kOff;
    const float* Wp = W + (size_t)(nBase + r16) * ldw + kOff;

    v8f acc = {};
    for (int kk = 0; kk < K; kk += 4) {
        v2f a = *(const v2f*)(Ap + kk);
        v2f b = *(const v2f*)(Wp + kk);
        acc = __builtin_amdgcn_wmma_f32_16x16x4_f32(
            false, a, false, b, (short)0, acc, false, false);
    }

    const int col = nBase + r16;
    const float bv = bias ? bias[col] : 0.0f;
#pragma unroll
    for (int i = 0; i < 8; ++i) {
        const int row = mBase + i + rowHi;
        float v = acc[i] + bv;
        if (accumulate) v += C[(size_t)row * N + col];
        C[(size_t)row * N + col] = v;
    }
}

// ---------------------------------------------------------------------------
// ControlUnit attention: scores over S=32, softmax, c_i = attn . cws
// grid = B, block = 256
// ---------------------------------------------------------------------------
__global__ __launch_bounds__(256) void control_attn(
    const float* __restrict__ cq,   // [B,D]
    const float* __restrict__ cws,  // [B,D,32]
    const float* __restrict__ w_ca, // [D]
    const float* __restrict__ b_ca, // [1]
    float* __restrict__ c_i)        // [B,D]
{
    const int b = blockIdx.x;
    const int tid = threadIdx.x;
    __shared__ float u[D];
    __shared__ float sc[32];

    for (int d = tid; d < D; d += 256) u[d] = cq[b * D + d] * w_ca[d];
    __syncthreads();

    const int wave = tid >> 5, lane = tid & 31;
#pragma unroll
    for (int j = 0; j < 4; ++j) {
        const int s = wave * 4 + j;
        float sum = 0.0f;
        for (int d = lane; d < D; d += 32)
            sum += u[d] * cws[((size_t)b * D + d) * 32 + s];
        for (int o = 16; o; o >>= 1) sum += __shfl_xor(sum, o, 32);
        if (lane == 0) sc[s] = sum + b_ca[0];
    }
    __syncthreads();

    if (tid < 32) {
        float v = sc[tid];
        float mx = v;
        for (int o = 16; o; o >>= 1) mx = fmaxf(mx, __shfl_xor(mx, o, 32));
        float e = __expf(v - mx);
        float s = e;
        for (int o = 16; o; o >>= 1) s += __shfl_xor(s, o, 32);
        sc[tid] = e / s;
    }
    __syncthreads();

    for (int d = tid; d < D; d += 256) {
        const float* row = &cws[((size_t)b * D + d) * 32];
        float acc = 0.0f;
#pragma unroll
        for (int s = 0; s < 32; ++s) acc += sc[s] * row[s];
        c_i[b * D + d] = acc;
    }
}

// ---------------------------------------------------------------------------
// Fused ReadUnit: per (batch, 16-row tile of HW):
//   KB2 = KBp @ w_kb^T ; I = mp * (KB2+b_kb) ; Ic = [I|KBp] @ w_mg^T + b_mg
//   mv[r] = sum_d c_i[d]*w_ra[d]*Ic[r,d] + b_ra ; m_new[b] += mv . KBp
// grid = (13, B), block = 256 (8 waves x 4 N-tiles each).
// KBp tile staged into LDS transposed ([dd][hw]) by the Tensor Data Mover:
//   tile x = 16 contiguous hw floats, y = 512 dd rows, stride 196; TDM
//   zero-fills past tensor_dim0 (ragged last tile) and inserts 2 pad DWORDs
//   every 16 DWORDs -> LDS row stride 18 (conflict-free).
// ---------------------------------------------------------------------------
#define LP 514  // padded LDS row stride for sI
#define RS 18   // padded LDS row stride for sKBt ([dd][r], 16 + 2 pad)

__global__ __launch_bounds__(256) void read_fused(
    const float* __restrict__ KB,   // [B, D, 196]
    const float* __restrict__ w_kb, // [D, D]
    const float* __restrict__ b_kb, // [D]
    const float* __restrict__ w_mg, // [D, 2D]
    const float* __restrict__ b_mg, // [D]
    const float* __restrict__ w_ra, // [D]
    const float* __restrict__ b_ra, // [1]
    const float* __restrict__ mp,   // [B, D]
    const float* __restrict__ c_i,  // [B, D]
    float* __restrict__ m_new)      // [B, D], pre-zeroed
{
    const int b   = blockIdx.y;
    const int hw0 = blockIdx.x * 16;
    const int tid = threadIdx.x;
    const int wave = tid >> 5, lane = tid & 31;

    __shared__ float sKBt[D * RS];   // sKBt[dd*RS + r] = KBp[r][dd]
    __shared__ float sI [16 * LP];   // sI[r*LP + col]  = I[r][col]
    __shared__ float sMV[16];

    if (tid < 16) sMV[tid] = b_ra[0];

#ifdef USE_TDM
    if (wave == 0) {
        // ---- D# group 0: count=1, lds_addr, global_addr, type=2 ----
        const uint64_t ga = (uint64_t)(uintptr_t)(KB + (size_t)b * D * NHW + hw0);
        const uint32_t la = (uint32_t)(uintptr_t)&sKBt[0];
        v4u g0;
        g0.x = 1u;                                   // count=1, is_restore=0
        g0.y = la;                                   // lds_addr
        g0.z = (uint32_t)ga;                         // global_addr[31:0]
        g0.w = ((uint32_t)(ga >> 32) & 0x01FFFFFFu)  // global_addr[56:32]
             | (2u << 30);                           // type=2 (image)
        // ---- D# group 1 ----
        union { uint32_t w[8]; v8i v; } g1;
        g1.w[0] = (2u << 16)    // data_size = 2 (4 bytes)
                | (1u << 20)    // pad_enable
                | (3u << 22)    // pad_interval = 16 DWORDs
                | (1u << 25);   // pad_amount   = 2 DWORDs  -> row stride 18
        g1.w[1] = (uint32_t)NHW << 16;   // tensor_dim0[15:0] (= 196)
        g1.w[2] = (uint32_t)D << 16;     // tensor_dim0 hi=0 | tensor_dim1[15:0]
        g1.w[3] = 16u << 16;             // tensor_dim1 hi=0 | tile_dim0 = 16
        g1.w[4] = (uint32_t)D;           // tile_dim1 = 512, tile_dim2 = 0
        g1.w[5] = (uint32_t)NHW;         // tensor_dim0_stride[31:0] = 196
        g1.w[6] = 0u;                    // stride0 hi | tensor_dim1_stride lo
        g1.w[7] = 0u;
        const v4i gz4 = {0, 0, 0, 0};
        const v8i gz8 = {0, 0, 0, 0, 0, 0, 0, 0};
        __builtin_amdgcn_tensor_load_to_lds(g0, g1.v, gz4, gz4, gz8, 0);
        __builtin_amdgcn_s_wait_tensorcnt(0);
    }
#else
    for (int i = tid; i < 16 * D; i += 256) {
        const int dd = i >> 4, r = i & 15;
        const int hw = hw0 + r;
        sKBt[dd * RS + r] = (hw < NHW) ? KB[((size_t)b * D + dd) * NHW + hw] : 0.0f;
    }
#endif
    __syncthreads();

    const int r16   = lane & 15;
    const int kOff  = (lane >> 4) * 2;
    const int rowHi = (lane >> 4) * 8;
    const int nBase0 = wave * 64;

    // Warm L2 for this wave's weight rows.
    __builtin_prefetch(&w_kb[(size_t)nBase0 * D], 0, 1);
    __builtin_prefetch(&w_mg[(size_t)nBase0 * 2 * D], 0, 1);

    v8f acc[4];

    // ---- GEMM1: KB2 tile = KBp @ w_kb^T, 4 N-tiles per wave ----
#pragma unroll
    for (int t = 0; t < 4; ++t) acc[t] = (v8f){};
    {
        const float* a0 = &sKBt[r16];
        for (int kk = 0; kk < D; kk += 4) {
            const int k0 = kk + kOff;
            v2f a;
            a.x = a0[k0 * RS];
            a.y = a0[(k0 + 1) * RS];
#pragma unroll
            for (int t = 0; t < 4; ++t) {
                v2f w = *(const v2f*)&w_kb[(size_t)(nBase0 + t * 16 + r16) * D + kOff + kk];
                acc[t] = __builtin_amdgcn_wmma_f32_16x16x4_f32(
                    false, a, false, w, (short)0, acc[t], false, false);
            }
        }
    }
    // I = mp[b,col] * (KB2 + b_kb[col]) -> LDS
#pragma unroll
    for (int t = 0; t < 4; ++t) {
        const int col = nBase0 + t * 16 + r16;
        const float scale = mp[b * D + col];
        const float bias  = b_kb[col];
#pragma unroll
        for (int i = 0; i < 8; ++i)
            sI[(i + rowHi) * LP + col] = scale * (acc[t][i] + bias);
    }
    __syncthreads();

    // ---- GEMM2: Ic tile = [I | KBp] @ w_mg^T (K split 512+512) ----
#pragma unroll
    for (int t = 0; t < 4; ++t) acc[t] = (v8f){};
    {
        const float* aI = &sI[r16 * LP + kOff];
        for (int kk = 0; kk < D; kk += 4) {
            v2f a = *(const v2f*)(aI + kk);
#pragma unroll
            for (int t = 0; t < 4; ++t) {
                v2f w = *(const v2f*)&w_mg[(size_t)(nBase0 + t * 16 + r16) * (2 * D) + kOff + kk];
                acc[t] = __builtin_amdgcn_wmma_f32_16x16x4_f32(
                    false, a, false, w, (short)0, acc[t], false, false);
            }
        }
        const float* aK = &sKBt[r16];
        for (int kk = 0; kk < D; kk += 4) {
            const int k0 = kk + kOff;
            v2f a;
            a.x = aK[k0 * RS];
            a.y = aK[(k0 + 1) * RS];
#pragma unroll
            for (int t = 0; t < 4; ++t) {
                v2f w = *(const v2f*)&w_mg[(size_t)(nBase0 + t * 16 + r16) * (2 * D) + D + kOff + kk];
                acc[t] = __builtin_amdgcn_wmma_f32_16x16x4_f32(
                    false, a, false, w, (short)0, acc[t], false, false);
            }
        }
    }
    // mv partials: reduce c_i[col]*w_ra[col]*(Ic+b_mg[col]) over columns.
#pragma unroll
    for (int t = 0; t < 4; ++t) {
        const int col = nBase0 + t * 16 + r16;
        const float cw   = c_i[b * D + col] * w_ra[col];
        const float bias = b_mg[col];
#pragma unroll
        for (int i = 0; i < 8; ++i) {
            float contrib = cw * (acc[t][i] + bias);
#pragma unroll
            for (int o = 1; o < 16; o <<= 1) contrib += __shfl_xor(contrib, o, 32);
            if (r16 == 0) atomicAdd(&sMV[i + rowHi], contrib);
        }
    }
    __syncthreads();

    // m_new[b,dd] += sum_r mv[r] * KBp[r][dd]   (zero rows contribute zero)
    for (int dd = tid; dd < D; dd += 256) {
        const float* row = &sKBt[dd * RS];
        float s = 0.0f;
#pragma unroll
        for (int r = 0; r < 16; ++r) s += sMV[r] * row[r];
        atomicAdd(&m_new[b * D + dd], s);
    }
}

// ---------------------------------------------------------------------------
// WriteUnit self-attention over P=3 previous controls. grid=B, block=96.
// ---------------------------------------------------------------------------
__global__ __launch_bounds__(96) void write_selfattn(
    const float* __restrict__ c_i,  // [B,D]
    const float* __restrict__ ls,   // [3,B,D]
    const float* __restrict__ w_sa, // [D]
    const float* __restrict__ b_sa, // [1]
    float* __restrict__ m_sa)       // [B,D]
{
    const int b = blockIdx.x, tid = threadIdx.x;
    const int p = tid >> 5, lane = tid & 31;
    __shared__ float sa[3];

    const float* lp = ls + ((size_t)p * 128 + b) * D;
    float sum = 0.0f;
    for (int d = lane; d < D; d += 32) sum += c_i[b * D + d] * lp[d] * w_sa[d];
    for (int o = 16; o; o >>= 1) sum += __shfl_xor(sum, o, 32);
    if (lane == 0) sa[p] = sum + b_sa[0];
    __syncthreads();

    for (int d = tid; d < D; d += 96) {
        m_sa[b * D + d] = sa[0] * ls[(size_t)(0 * 128 + b) * D + d]
                        + sa[1] * ls[(size_t)(1 * 128 + b) * D + d]
                        + sa[2] * ls[(size_t)(2 * 128 + b) * D + d];
    }
}

// ---------------------------------------------------------------------------
// Memory gate: m = g*m_prev + (1-g)*m2, g = sigmoid(glin)
// ---------------------------------------------------------------------------
__global__ __launch_bounds__(256) void gate_out(
    const float* __restrict__ glin, const float* __restrict__ m_prev,
    const float* __restrict__ m2, float* __restrict__ m_out)
{
    const int i = blockIdx.x * 256 + threadIdx.x;
    const float g = 1.0f / (1.0f + __expf(-glin[i]));
    m_out[i] = g * m_prev[i] + (1.0f - g) * m2[i];
}

// ---------------------------------------------------------------------------
extern "C" void kernel_launch(void* const* d_in, const int* in_sizes, int n_in,
                              void* d_out, int out_size, void* d_ws, size_t ws_size,
                              hipStream_t stream) {
    const float* ls_c_i = (const float*)d_in[0];
    const float* m_prev = (const float*)d_in[1];
    const float* q      = (const float*)d_in[2];
    const float* cws    = (const float*)d_in[3];
    const float* KB     = (const float*)d_in[4];
    const float* w_q  = (const float*)d_in[5];  const float* b_q  = (const float*)d_in[6];
    const float* w_ct = (const float*)d_in[7];  const float* b_ct = (const float*)d_in[8];
    const float* w_ca = (const float*)d_in[9];  const float* b_ca = (const float*)d_in[10];
    const float* w_mp = (const float*)d_in[11]; const float* b_mp = (const float*)d_in[12];
    const float* w_kb = (const float*)d_in[13]; const float* b_kb = (const float*)d_in[14];
    const float* w_mg = (const float*)d_in[15]; const float* b_mg = (const float*)d_in[16];
    const float* w_ra = (const float*)d_in[17]; const float* b_ra = (const float*)d_in[18];
    const float* w_wm = (const float*)d_in[19]; const float* b_wm = (const float*)d_in[20];
    const float* w_sa = (const float*)d_in[21]; const float* b_sa = (const float*)d_in[22];
    const float* w_am = (const float*)d_in[23]; const float* b_am = (const float*)d_in[24];
    const float* w_g  = (const float*)d_in[25]; const float* b_g  = (const float*)d_in[26];

    const int BD = 128 * D;                      // 65536
    float* ws     = (float*)d_ws;
    float* q_i    = ws + 0 * BD;
    float* cq     = ws + 1 * BD;
    float* mp     = ws + 2 * BD;
    float* m_new  = ws + 3 * BD;
    float* m1     = ws + 4 * BD;
    float* msa    = ws + 5 * BD;
    float* m2     = ws + 6 * BD;
    float* glin   = ws + 7 * BD;

    float* c_i_out = (float*)d_out;              // output 0: c_i [128,512]
    float* m_out   = (float*)d_out + BD;         // output 1: m   [128,512]
    const float* c_prev = ls_c_i + 2 * BD;       // ls_c_i[-1]

    const dim3 gT(D / 16, 128 / 16), bW(32);

    // ---- ControlUnit ----
    gemm_xwT<<<gT, bW, 0, stream>>>(q,      w_q,        b_q,  q_i, D, D, D, D,     0);
    gemm_xwT<<<gT, bW, 0, stream>>>(q_i,    w_ct,       b_ct, cq,  D, D, D, 2 * D, 0);
    gemm_xwT<<<gT, bW, 0, stream>>>(c_prev, w_ct + D,   NULL, cq,  D, D, D, 2 * D, 1);
    control_attn<<<128, 256, 0, stream>>>(cq, cws, w_ca, b_ca, c_i_out);

    // ---- ReadUnit ----
    gemm_xwT<<<gT, bW, 0, stream>>>(m_prev, w_mp, b_mp, mp, D, D, D, D, 0);
    (void)hipMemsetAsync(m_new, 0, (size_t)BD * sizeof(float), stream);
    read_fused<<<dim3(13, 128), 256, 0, stream>>>(KB, w_kb, b_kb, w_mg, b_mg,
                                                  w_ra, b_ra, mp, c_i_out, m_new);

    // ---- WriteUnit ----
    gemm_xwT<<<gT, bW, 0, stream>>>(m_new,  w_wm,     b_wm, m1, D, D, D, 2 * D, 0);
    gemm_xwT<<<gT, bW, 0, stream>>>(m_prev, w_wm + D, NULL, m1, D, D, D, 2 * D, 1);
    write_selfattn<<<128, 96, 0, stream>>>(c_i_out, ls_c_i, w_sa, b_sa, msa);
    gemm_xwT<<<gT, bW, 0, stream>>>(msa, w_am,     b_am, m2, D, D, D, 2 * D, 0);
    gemm_xwT<<<gT, bW, 0, stream>>>(m1,  w_am + D, NULL, m2, D, D, D, 2 * D, 1);
    gemm_xwT<<<gT, bW, 0, stream>>>(c_i_out, w_g, b_g, glin, D, D, D, D, 0);
    gate_out<<<BD / 256, 256, 0, stream>>>(glin, m_prev, m2, m_out);
}